// DeformableConv2d_5781025980835
// MI455X (gfx1250) — compile-verified
//
#include <hip/hip_runtime.h>

typedef __attribute__((ext_vector_type(2))) float v2f;
typedef __attribute__((ext_vector_type(8))) float v8f;

#define BATCH 8
#define CIN   64
#define HH    128
#define WW    128
#define HWSZ  (HH * WW)      // 16384
#define KTAP  9
#define OOUT  64
#define FDIM  (KTAP * CIN)   // 576
#define OFFC  18

// ---------------------------------------------------------------------------
// Kernel 1: 3x3 conv (64 -> 18 channels), stride 1, pad 1, NCHW.
// One thread per pixel; offset weights cached in LDS (18*576*4 = 41.5 KB).
// Output: off[b][18][h][w] in workspace.
// ---------------------------------------------------------------------------
__global__ __launch_bounds__(128)
void dc_offset_conv(const float* __restrict__ x,
                    const float* __restrict__ ow,
                    const float* __restrict__ ob,
                    float* __restrict__ off) {
    __shared__ float sw[OFFC * FDIM];
    const int tid = threadIdx.x;
    for (int i = tid; i < OFFC * FDIM; i += 128) sw[i] = ow[i];
    __syncthreads();

    const int h = blockIdx.x;
    const int b = blockIdx.y;
    const int w = tid;

    float acc[OFFC];
#pragma unroll
    for (int j = 0; j < OFFC; ++j) acc[j] = ob[j];

    const float* xb = x + (size_t)b * CIN * HWSZ;
    for (int c = 0; c < CIN; ++c) {
        float v[9];
#pragma unroll
        for (int dy = 0; dy < 3; ++dy) {
            const int yy = h + dy - 1;
            const bool yok = (yy >= 0) && (yy < HH);
#pragma unroll
            for (int dx = 0; dx < 3; ++dx) {
                const int xx = w + dx - 1;
                const bool ok = yok && (xx >= 0) && (xx < WW);
                v[dy * 3 + dx] = ok ? xb[(size_t)c * HWSZ + yy * WW + xx] : 0.0f;
            }
        }
#pragma unroll
        for (int j = 0; j < OFFC; ++j) {
            const float* wp = &sw[j * FDIM + c * 9];
            float s = acc[j];
#pragma unroll
            for (int t = 0; t < 9; ++t) s = fmaf(v[t], wp[t], s);
            acc[j] = s;
        }
    }
    const size_t base = (size_t)b * OFFC * HWSZ + (size_t)h * WW + w;
#pragma unroll
    for (int j = 0; j < OFFC; ++j) off[base + (size_t)j * HWSZ] = acc[j];
}

// ---------------------------------------------------------------------------
// Kernel 2: fused bilinear gather + fp32 WMMA contraction.
// WG = 256 threads (8 wave32), one image row (128 pixels) per (row, batch).
// GEMM: out[o][p] = sum_f wgt[o*576 + f] * samp[f][p], f = k*64 + c
// (the reference's "faithful mismatch" makes the weight access linear in f).
// M = out channel (64, 4 tiles of 16), N = pixel (128, 8 subtiles of 16),
// K = f (576, 9 taps x 16 steps of 4) via V_WMMA_F32_16X16X4_F32.
// ---------------------------------------------------------------------------
#define LDSTR 144   // 2*144 mod 64 = 32 -> lane halves hit disjoint banks

__global__ __launch_bounds__(256)
void dc_deform_wmma(const float* __restrict__ x,
                    const float* __restrict__ wgt,
                    const float* __restrict__ bias,
                    const float* __restrict__ off,
                    float* __restrict__ out) {
    __shared__ float As[CIN * LDSTR];   // 64 x 144 floats = 36.9 KB

    const int tid  = threadIdx.x;
    const int h    = blockIdx.x;
    const int b    = blockIdx.y;
    const int p    = tid & 127;         // pixel in row (gather role)
    const int cgrp = tid >> 7;          // channel group 0..1 (gather role)
    const int lane = tid & 31;
    const int wv   = tid >> 5;          // wave id 0..7
    const int mt   = wv & 3;            // M tile (out-chan base = mt*16)
    const int ng   = wv >> 2;           // N group: subtiles ng*4 .. ng*4+3

    v8f acc[4] = {};

    const float* xb = x + (size_t)b * CIN * HWSZ;
    const size_t offbase = (size_t)b * OFFC * HWSZ + (size_t)h * WW + p;

    const int nlo  = lane & 15;
    const int half = lane >> 4;

    for (int k = 0; k < KTAP; ++k) {
        // ---- per-pixel bilinear meta (depends only on (b,k,h,p)) ----
        const float offx = off[offbase + (size_t)k * HWSZ];
        const float offy = off[offbase + (size_t)(KTAP + k) * HWSZ];
        const float px = (float)(k % 3 - 1) + offx;
        const float py = (float)(k / 3 - 1) + offy;
        const float x0f = floorf(px), y0f = floorf(py);
        const float dx = px - x0f, dy = py - y0f;
        const int x0 = (int)x0f, y0 = (int)y0f;
        float w00 = (1.0f - dx) * (1.0f - dy);
        float w10 = dx * (1.0f - dy);
        float w01 = (1.0f - dx) * dy;
        float w11 = dx * dy;
        const float vx0 = (x0 >= 0 && x0 < WW) ? 1.0f : 0.0f;
        const float vx1 = (x0 + 1 >= 0 && x0 + 1 < WW) ? 1.0f : 0.0f;
        const float vy0 = (y0 >= 0 && y0 < HH) ? 1.0f : 0.0f;
        const float vy1 = (y0 + 1 >= 0 && y0 + 1 < HH) ? 1.0f : 0.0f;
        w00 *= vx0 * vy0; w10 *= vx1 * vy0; w01 *= vx0 * vy1; w11 *= vx1 * vy1;
        const int x0c = min(max(x0, 0), WW - 1);
        const int x1c = min(max(x0 + 1, 0), WW - 1);
        const int y0c = min(max(y0, 0), HH - 1);
        const int y1c = min(max(y0 + 1, 0), HH - 1);
        const int i00 = y0c * WW + x0c, i10 = y0c * WW + x1c;
        const int i01 = y1c * WW + x0c, i11 = y1c * WW + x1c;

        __syncthreads();   // previous tap's LDS tile fully consumed
        // ---- gather 32 channels for this pixel into LDS ----
#pragma unroll 4
        for (int cc = 0; cc < 32; ++cc) {
            const int c = cgrp * 32 + cc;
            const float* xc = xb + (size_t)c * HWSZ;
            const float v = w00 * xc[i00] + w10 * xc[i10]
                          + w01 * xc[i01] + w11 * xc[i11];
            As[c * LDSTR + p] = v;
        }
        __syncthreads();

        // ---- per-tap A fragments (weights), ISA 16x4 f32 A layout:
        //      lane.x = K = 2*half + 0 , lane.y = K = 2*half + 1 ----
        v2f wf[16];
        {
            const int m = mt * 16 + nlo;
            const float* wrow = wgt + (size_t)m * FDIM + k * CIN + 2 * half;
#pragma unroll
            for (int s = 0; s < 16; ++s) {
                wf[s].x = wrow[s * 4 + 0];
                wf[s].y = wrow[s * 4 + 1];
            }
        }

        // ---- 4 N-subtiles x 16 K-steps of WMMA ----
#pragma unroll
        for (int t = 0; t < 4; ++t) {
            const int pcol = (ng * 4 + t) * 16 + nlo;
#pragma unroll
            for (int s = 0; s < 16; ++s) {
                const int crow = s * 4 + 2 * half;   // B 4x16 layout mirror
                v2f bf;
                bf.x = As[(crow + 0) * LDSTR + pcol];
                bf.y = As[(crow + 1) * LDSTR + pcol];
                acc[t] = __builtin_amdgcn_wmma_f32_16x16x4_f32(
                    false, wf[s], false, bf, (short)0, acc[t], false, false);
            }
        }
    }

    // ---- epilogue: bias + coalesced row-major stores (D layout) ----
    float bv[8];
#pragma unroll
    for (int r = 0; r < 8; ++r) bv[r] = bias[mt * 16 + r + 8 * half];

#pragma unroll
    for (int t = 0; t < 4; ++t) {
        const int pcol = (ng * 4 + t) * 16 + nlo;
#pragma unroll
        for (int r = 0; r < 8; ++r) {
            const int m = mt * 16 + r + 8 * half;
            out[((size_t)(b * OOUT + m)) * HWSZ + (size_t)h * WW + pcol] =
                acc[t][r] + bv[r];
        }
    }
}

// ---------------------------------------------------------------------------
extern "C" void kernel_launch(void* const* d_in, const int* in_sizes, int n_in,
                              void* d_out, int out_size, void* d_ws, size_t ws_size,
                              hipStream_t stream) {
    (void)in_sizes; (void)n_in; (void)out_size; (void)ws_size;
    const float* x        = (const float*)d_in[0];
    const float* weight   = (const float*)d_in[1];
    const float* bias     = (const float*)d_in[2];
    const float* offset_w = (const float*)d_in[3];
    const float* offset_b = (const float*)d_in[4];
    float* out = (float*)d_out;
    float* off = (float*)d_ws;   // 8*18*16384 floats = 9.4 MB scratch

    dc_offset_conv<<<dim3(HH, BATCH), 128, 0, stream>>>(x, offset_w, offset_b, off);
    dc_deform_wmma<<<dim3(HH, BATCH), 256, 0, stream>>>(x, weight, bias, off, out);
}